// CondNet_17016660427311
// MI455X (gfx1250) — compile-verified
//
#include <hip/hip_runtime.h>

typedef __attribute__((ext_vector_type(2))) float v2f;
typedef __attribute__((ext_vector_type(4))) float v4f;
typedef __attribute__((ext_vector_type(8))) float v8f;

#define B_DIM   512
#define NUM_IN  1024
#define NUM_MID 8192
#define NUM_OUT 1000
#define FAN_IN  64

// ---------------------------------------------------------------------------
// Stage 1: H1t[m][b] = relu( sum_k W_in[m][k] * x[b][k] + b_in[m] )
// C(M=8192, N=512) = W_in(8192x1024) * x^T(1024x512), output stored transposed
// (flat [mid][batch]). One wave computes a 16(M) x 64(N) strip; the A fragment
// is reused across 4 WMMAs.
// ---------------------------------------------------------------------------
__global__ __launch_bounds__(256) void gemm_in_wmma(
    const float* __restrict__ x,     // [512][1024]
    const float* __restrict__ W_in,  // [8192][1024]
    const float* __restrict__ b_in,  // [8192]
    float* __restrict__ Ht)          // [8192][512]
{
    const int lane  = threadIdx.x & 31;
    const int wave  = threadIdx.x >> 5;
    const int gwave = blockIdx.x * 8 + wave;      // 512 blocks * 8 waves = 4096
    const int m0 = (gwave >> 3) * 16;             // 512 M-tiles
    const int n0 = (gwave & 7) * 64;              // 8 N-groups of 64
    const int r  = lane & 15;                     // row (A) / col (B,C)
    const int kh = lane >> 4;                     // K half: 0 -> K{0,1}, 1 -> K{2,3}
    const int kp = kh * 2;

    const float* arow  = W_in + (size_t)(m0 + r) * NUM_IN + kp;
    const float* brow0 = x + (size_t)(n0 +  0 + r) * NUM_IN + kp;
    const float* brow1 = x + (size_t)(n0 + 16 + r) * NUM_IN + kp;
    const float* brow2 = x + (size_t)(n0 + 32 + r) * NUM_IN + kp;
    const float* brow3 = x + (size_t)(n0 + 48 + r) * NUM_IN + kp;

    v8f acc0 = {}, acc1 = {}, acc2 = {}, acc3 = {};

#pragma unroll 4
    for (int k = 0; k < NUM_IN; k += 4) {
        v2f a  = *(const v2f*)(arow  + k);
        v2f f0 = *(const v2f*)(brow0 + k);
        v2f f1 = *(const v2f*)(brow1 + k);
        v2f f2 = *(const v2f*)(brow2 + k);
        v2f f3 = *(const v2f*)(brow3 + k);
        acc0 = __builtin_amdgcn_wmma_f32_16x16x4_f32(false, a, false, f0, (short)0, acc0, false, false);
        acc1 = __builtin_amdgcn_wmma_f32_16x16x4_f32(false, a, false, f1, (short)0, acc1, false, false);
        acc2 = __builtin_amdgcn_wmma_f32_16x16x4_f32(false, a, false, f2, (short)0, acc2, false, false);
        acc3 = __builtin_amdgcn_wmma_f32_16x16x4_f32(false, a, false, f3, (short)0, acc3, false, false);
    }

#pragma unroll
    for (int v = 0; v < 8; ++v) {
        const int m = m0 + v + kh * 8;            // C row for this VGPR/lane-half
        const float bias = b_in[m];
        float* orow = Ht + (size_t)m * B_DIM;
        orow[n0 +  0 + r] = fmaxf(acc0[v] + bias, 0.0f);
        orow[n0 + 16 + r] = fmaxf(acc1[v] + bias, 0.0f);
        orow[n0 + 32 + r] = fmaxf(acc2[v] + bias, 0.0f);
        orow[n0 + 48 + r] = fmaxf(acc3[v] + bias, 0.0f);
    }
}

// ---------------------------------------------------------------------------
// Condensed layer (transposed): out[o][b] = relu( sum_f Ht_in[idx[o][f]][b]
//                                                 * W[o][f] + bm[o] )
// One 128-thread block per neuron o; each thread owns 4 batch columns (v4f),
// so each wave issues one contiguous 512B burst per fan-in element. idx/W
// loads are block-uniform (scalar path).
// INTERLEAVE=0: flat [o][b] output (feeds condensed layer 1).
// INTERLEAVE=1: K-pair layout Hp[o>>1][b][o&1] (feeds stage-4 WMMA B-operand
//               as contiguous v2f fragments).
// ---------------------------------------------------------------------------
template <int INTERLEAVE>
__global__ __launch_bounds__(128) void condensed_layer(
    const float* __restrict__ Ht_in,  // [8192][512]
    const float* __restrict__ Wm,     // [8192][64]
    const float* __restrict__ bm,     // [8192]
    const int*   __restrict__ idx,    // [8192][64]
    float* __restrict__ Ht_out)       // [8192][512] flat or pair-interleaved
{
    const int o  = blockIdx.x;
    const int b4 = threadIdx.x * 4;   // 128 threads * 4 = 512 batch columns
    const int*   io = idx + (size_t)o * FAN_IN;
    const float* wo = Wm  + (size_t)o * FAN_IN;

    v4f acc = {};
#pragma unroll 8
    for (int f = 0; f < FAN_IN; ++f) {
        const int   col = io[f];
        const float w   = wo[f];
        v4f h = *(const v4f*)(Ht_in + (size_t)col * B_DIM + b4);
        acc += h * w;
    }
    const float bias = bm[o];
    const float r0 = fmaxf(acc.x + bias, 0.0f);
    const float r1 = fmaxf(acc.y + bias, 0.0f);
    const float r2 = fmaxf(acc.z + bias, 0.0f);
    const float r3 = fmaxf(acc.w + bias, 0.0f);

    if (INTERLEAVE) {
        // element (o, b) -> Hp[(o>>1)*1024 + 2*b + (o&1)]
        float* op = Ht_out + (size_t)(o >> 1) * (2 * B_DIM) + 2 * b4 + (o & 1);
        op[0] = r0; op[2] = r1; op[4] = r2; op[6] = r3;
    } else {
        v4f outv; outv.x = r0; outv.y = r1; outv.z = r2; outv.w = r3;
        *(v4f*)(Ht_out + (size_t)o * B_DIM + b4) = outv;
    }
}

// ---------------------------------------------------------------------------
// Stage 4: out[b][j] = sum_m H3[b][m] * W_out[j][m] + b_out[j]
// Computed as C(J=1000, N=512) = W_out(1000x8192) * H3t(8192x512), with H3t in
// K-pair-interleaved form Hp[k>>1][n][k&1], so each B fragment (B[k+kp][n],
// B[k+kp+1][n]) is one contiguous v2f -> 5 b64 loads per 4 WMMAs.
// j-tile 62 clamps its A-row loads (EXEC stays all-ones) and predicates stores.
// ---------------------------------------------------------------------------
__global__ __launch_bounds__(256) void gemm_out_wmma(
    const float* __restrict__ Hp,     // [4096][512][2] pair-interleaved H3t
    const float* __restrict__ W_out,  // [1000][8192]
    const float* __restrict__ b_out,  // [1000]
    float* __restrict__ out)          // [512][1000]
{
    const int lane = threadIdx.x & 31;
    const int wave = threadIdx.x >> 5;            // 8 waves -> 8 N-groups of 64
    const int m0 = blockIdx.x * 16;               // 63 blocks cover j=0..1007
    const int n0 = wave * 64;
    const int r  = lane & 15;
    const int kh = lane >> 4;
    const int kp = kh * 2;

    int jr = m0 + r;
    if (jr > NUM_OUT - 1) jr = NUM_OUT - 1;       // clamp, keep EXEC all-ones
    const float* arow = W_out + (size_t)jr * NUM_MID + kp;

    // B fragment base: row (k>>1)+kh of Hp, column pair at 2*(n0+r)
    const float* bbase = Hp + (size_t)kh * (2 * B_DIM) + 2 * (n0 + r);

    v8f acc0 = {}, acc1 = {}, acc2 = {}, acc3 = {};

#pragma unroll 4
    for (int k = 0; k < NUM_MID; k += 4) {
        v2f a = *(const v2f*)(arow + k);
        const float* bp = bbase + (size_t)(k >> 1) * (2 * B_DIM);
        v2f f0 = *(const v2f*)(bp +  0);          // n0 +  0 + r
        v2f f1 = *(const v2f*)(bp + 32);          // n0 + 16 + r
        v2f f2 = *(const v2f*)(bp + 64);          // n0 + 32 + r
        v2f f3 = *(const v2f*)(bp + 96);          // n0 + 48 + r
        acc0 = __builtin_amdgcn_wmma_f32_16x16x4_f32(false, a, false, f0, (short)0, acc0, false, false);
        acc1 = __builtin_amdgcn_wmma_f32_16x16x4_f32(false, a, false, f1, (short)0, acc1, false, false);
        acc2 = __builtin_amdgcn_wmma_f32_16x16x4_f32(false, a, false, f2, (short)0, acc2, false, false);
        acc3 = __builtin_amdgcn_wmma_f32_16x16x4_f32(false, a, false, f3, (short)0, acc3, false, false);
    }

#pragma unroll
    for (int v = 0; v < 8; ++v) {
        const int j = m0 + v + kh * 8;
        if (j < NUM_OUT) {                         // store predicate only
            const float bias = b_out[j];
            out[(size_t)(n0 +  0 + r) * NUM_OUT + j] = acc0[v] + bias;
            out[(size_t)(n0 + 16 + r) * NUM_OUT + j] = acc1[v] + bias;
            out[(size_t)(n0 + 32 + r) * NUM_OUT + j] = acc2[v] + bias;
            out[(size_t)(n0 + 48 + r) * NUM_OUT + j] = acc3[v] + bias;
        }
    }
}

extern "C" void kernel_launch(void* const* d_in, const int* in_sizes, int n_in,
                              void* d_out, int out_size, void* d_ws, size_t ws_size,
                              hipStream_t stream)
{
    const float* x     = (const float*)d_in[0];
    const float* W_in  = (const float*)d_in[1];
    const float* b_in  = (const float*)d_in[2];
    const float* W_mid = (const float*)d_in[3];   // [2][8192][64]
    const float* b_mid = (const float*)d_in[4];   // [2][8192]
    const float* W_out = (const float*)d_in[5];
    const float* b_out = (const float*)d_in[6];
    const int*   idx   = (const int*)d_in[7];     // [8192][64]
    float* out = (float*)d_out;

    float* Ha = (float*)d_ws;                             // [8192][512]
    float* Hb = Ha + (size_t)NUM_MID * B_DIM;             // [8192][512]

    // Stage 1: H1t = relu(W_in * x^T + b_in)            (flat layout -> Ha)
    gemm_in_wmma<<<dim3(512), dim3(256), 0, stream>>>(x, W_in, b_in, Ha);

    // Stage 2: condensed layer 0 (Ha flat -> Hb flat)
    condensed_layer<0><<<dim3(NUM_MID), dim3(128), 0, stream>>>(
        Ha, W_mid, b_mid, idx, Hb);

    // Stage 3: condensed layer 1 (Hb flat -> Ha pair-interleaved)
    condensed_layer<1><<<dim3(NUM_MID), dim3(128), 0, stream>>>(
        Hb, W_mid + (size_t)NUM_MID * FAN_IN, b_mid + NUM_MID, idx, Ha);

    // Stage 4: out = H3 @ W_out^T + b_out
    gemm_out_wmma<<<dim3(63), dim3(256), 0, stream>>>(Ha, W_out, b_out, out);
}